// HSIC_1855425872455
// MI455X (gfx1250) — compile-verified
//
#include <hip/hip_runtime.h>
#include <hip/hip_bf16.h>

// HSIC for X[2048,16] f32 -> scalar f32.
//
// Math mapping:
//  sigma_c^2 = mean_{i,j}(x_i-x_j)^2 = 2*(E[x^2]-E[x]^2)   (column moments only)
//  Y[i,c]    = X[i,c]*sqrt(log2(e)/(2*sigma_c^2))  =>  K_c[i,j] = exp2(-(Yi-Yj)^2)
//  T[a,b]    = sum_{i,j} K_a[i,j]K_b[i,j]   -> 16x16 Gram over K-dim n^2 = WMMA f32 16x16x4
//  A[c,i]    = rowsums, S[c] = totals, Dm = A A^T (16x16 Gram over n=2048, also WMMA)
//
// Diagonal handling: the inner loop keeps the (un-zeroed) diagonal K[i,i]=exp2(0)=1
// and we correct analytically afterwards:  T -= n (every entry),  A[c,i] -= 1.
// This removes all v_cmp/v_cndmask from the hot loop.
//
// Workspace layout (needs 512 KB):
//   [0,128K)      Y  as [2048][16] f32
//   [128K,256K)   A  as [2048][16] f32 (diagonal-corrected rowsums)
//   [256K,512K)   Tpart as [256 blocks][256] f32 (per-block partial T, canonical m*16+n)

typedef float v2f __attribute__((ext_vector_type(2)));
typedef float v8f __attribute__((ext_vector_type(8)));

static __device__ __forceinline__ v8f wmma_gram(v2f ab, v8f c) {
    // D = A*B + C with A (16x4 f32) and B (4x16 f32) sharing the same registers:
    // per ISA f32 layouts, the Gram case (B == A^T) maps both operands onto
    // identical lane/VGPR contents.
    return __builtin_amdgcn_wmma_f32_16x16x4_f32(false, ab, false, ab,
                                                 (short)0, c, false, false);
}

// ---------------------------------------------------------------- prep ------
__global__ void hsic_prep(const float* __restrict__ X, float* __restrict__ Y) {
    __shared__ float ps[16][16], ps2[16][16], sq[16];
    const int t = threadIdx.x;        // 256 threads
    const int c = t & 15;             // feature column
    const int g = t >> 4;             // 16 row groups
    float s = 0.f, s2 = 0.f;
    for (int i = g; i < 2048; i += 16) {
        float x = X[i * 16 + c];
        s += x; s2 += x * x;
    }
    ps[c][g] = s; ps2[c][g] = s2;
    __syncthreads();
    if (t < 16) {
        float S = 0.f, S2 = 0.f;
        for (int gg = 0; gg < 16; ++gg) { S += ps[t][gg]; S2 += ps2[t][gg]; }
        const float inv_n = 1.f / 2048.f;
        float meanD = 2.f * (S2 * inv_n - (S * inv_n) * (S * inv_n)); // = sigma^2
        // exponent in log2 units: K = exp2(-(yi-yj)^2), y = x*sqrt(log2e/(2 meanD))
        sq[t] = sqrtf(1.44269504088896340736f / (2.f * meanD));
    }
    __syncthreads();
    const float m = sq[c];
    for (int i = g; i < 2048; i += 16)
        Y[i * 16 + c] = X[i * 16 + c] * m;
}

// ---------------------------------------------------------------- main ------
// 256 blocks x 256 threads (8 waves). Wave w of block b owns row i = b*8+w.
__global__ void __launch_bounds__(256) hsic_main(const float* __restrict__ Y,
                                                 float* __restrict__ A,
                                                 float* __restrict__ Tpart) {
    extern __shared__ float ldsY[];       // 2048*16 f32 = 128 KB dynamic
    __shared__ float red[8 * 256];        // 8 KB reduction buffer

    const int tid  = threadIdx.x;
    const int lane = tid & 31;
    const int wave = tid >> 5;
    const int c    = lane & 15;           // feature
    const int hi   = lane >> 4;           // lane half selects pair slots {0,1} vs {2,3}

    // Stage Y into LDS (all 8 waves share it)
    for (int k = tid; k < 2048 * 16; k += 256) ldsY[k] = Y[k];
    __syncthreads();

    const int   i  = blockIdx.x * 8 + wave;
    const float yi = ldsY[i * 16 + c];

    v8f acc0 = {};   // two accumulators -> two independent WMMA chains (ILP)
    v8f acc1 = {};
    float rs = 0.f;  // partial row sum for feature c over this lane-half's slots

    for (int j0 = 0; j0 < 2048; j0 += 8) {
        // group 0: pairs (i, j0 .. j0+3); this lane covers slots via lane half
        const float* p0 = &ldsY[(j0 + 2 * hi) * 16 + c];
        float d0 = yi - p0[0];
        float d1 = yi - p0[16];
        float k0 = __builtin_amdgcn_exp2f(-(d0 * d0));   // diag stays 1, fixed later
        float k1 = __builtin_amdgcn_exp2f(-(d1 * d1));

        // group 1: pairs (i, j0+4 .. j0+7)
        const float* p1 = p0 + 64;
        float d2 = yi - p1[0];
        float d3 = yi - p1[16];
        float k2 = __builtin_amdgcn_exp2f(-(d2 * d2));
        float k3 = __builtin_amdgcn_exp2f(-(d3 * d3));

        v2f ab0; ab0[0] = k0; ab0[1] = k1;
        v2f ab1; ab1[0] = k2; ab1[1] = k3;
        acc0 = wmma_gram(ab0, acc0);
        acc1 = wmma_gram(ab1, acc1);

        rs += (k0 + k1) + (k2 + k3);
    }

    // Row sum A[c,i]: combine both lane halves (wave32: xor with 16), then
    // subtract the diagonal's exp2(0)=1 contribution.
    rs += __shfl_xor(rs, 16, 32);
    if (lane < 16) A[i * 16 + c] = rs - 1.0f;

    // Merge the two accumulators and reduce 8 waves' 16x16 tiles through LDS.
#pragma unroll
    for (int r = 0; r < 8; ++r) red[wave * 256 + r * 32 + lane] = acc0[r] + acc1[r];
    __syncthreads();
    float tsum = 0.f;
#pragma unroll
    for (int w = 0; w < 8; ++w) tsum += red[w * 256 + tid];
    // slot tid = r*32 + lane  ->  m = r + 8*(lane>=16), n = lane&15 (C layout)
    const int m = (tid >> 5) + 8 * ((tid >> 4) & 1);
    const int n = tid & 15;
    Tpart[blockIdx.x * 256 + m * 16 + n] = tsum;
}

// -------------------------------------------------------------- finalize ----
__global__ void __launch_bounds__(256) hsic_final(const float* __restrict__ A,
                                                  const float* __restrict__ Tpart,
                                                  int nblocks,
                                                  float* __restrict__ out) {
    __shared__ float sT[256], sDm[256], sS[16], sP[16][16], red[8 * 256];
    const int tid  = threadIdx.x;
    const int lane = tid & 31;
    const int wave = tid >> 5;
    const int c    = tid & 15;
    const int g    = tid >> 4;
    const int hi   = lane >> 4;

    // 1) T = fixed-order reduce of per-block partials, then remove the n
    //    diagonal contributions (K_a[i,i]*K_b[i,i] = 1 for every (a,b)).
    float ts = 0.f;
    for (int b = 0; b < nblocks; ++b) ts += Tpart[b * 256 + tid];
    sT[tid] = ts - 2048.0f;

    // 2) S[c] partials (deterministic tree); A is already diagonal-corrected.
    float ss = 0.f;
    for (int i = g; i < 2048; i += 16) ss += A[i * 16 + c];
    sP[c][g] = ss;
    __syncthreads();
    if (tid < 16) {
        float acc = 0.f;
        for (int gg = 0; gg < 16; ++gg) acc += sP[tid][gg];
        sS[tid] = acc;
    }

    // 3) Dm = A A^T via the same WMMA f32 Gram; wave w covers i in [w*256,(w+1)*256)
    v8f accD = {};
    for (int i0 = wave * 256; i0 < (wave + 1) * 256; i0 += 4) {
        int ia = i0 + 2 * hi;
        v2f ab; ab[0] = A[ia * 16 + c]; ab[1] = A[(ia + 1) * 16 + c];
        accD = wmma_gram(ab, accD);
    }
#pragma unroll
    for (int r = 0; r < 8; ++r) red[wave * 256 + r * 32 + lane] = accD[r];
    __syncthreads();
    float dsum = 0.f;
#pragma unroll
    for (int w = 0; w < 8; ++w) dsum += red[w * 256 + tid];
    const int m = (tid >> 5) + 8 * ((tid >> 4) & 1);
    sDm[m * 16 + (tid & 15)] = dsum;
    __syncthreads();

    // 4) HSIC matrix + masked sum of squares
    const int a = tid >> 4, b2 = tid & 15;
    const float nn = 2048.f;
    const float c0 = 1.f / (nn * (nn - 3.f));
    const float f1 = 1.f / ((nn - 1.f) * (nn - 2.f));
    const float f2 = 2.f / (nn - 2.f);
    float h = c0 * (sT[tid] + sS[a] * sS[b2] * f1 - f2 * sDm[tid]);
    float v = (a < b2) ? h * h : 0.f;
    __syncthreads();
    red[tid] = v;
    __syncthreads();
    if (tid == 0) {
        float tot = 0.f;
        for (int k = 0; k < 256; ++k) tot += red[k];
        out[0] = tot;
    }
}

// ---------------------------------------------------------------------------
extern "C" void kernel_launch(void* const* d_in, const int* in_sizes, int n_in,
                              void* d_out, int out_size, void* d_ws, size_t ws_size,
                              hipStream_t stream) {
    (void)in_sizes; (void)n_in; (void)out_size; (void)ws_size;
    const float* X = (const float*)d_in[0];
    float* out = (float*)d_out;

    char*  ws    = (char*)d_ws;
    float* Y     = (float*)(ws);                // 128 KB
    float* A     = (float*)(ws + (128u << 10)); // 128 KB
    float* Tpart = (float*)(ws + (256u << 10)); // 256 KB

    hsic_prep<<<1, 256, 0, stream>>>(X, Y);
    hsic_main<<<256, 256, 2048 * 16 * sizeof(float), stream>>>(Y, A, Tpart);
    hsic_final<<<1, 256, 0, stream>>>(A, Tpart, 256, out);
}